// SAMLossMultiInstance_57543971832570
// MI455X (gfx1250) — compile-verified
//
#include <hip/hip_runtime.h>
#include <math.h>

// ---------------------------------------------------------------------------
// SAM multi-instance loss, fused single-pass streaming kernel for gfx1250.
//   s[b,i]    = sum_px softplus(pred[b,i,px])          (raw v_exp_f32/v_log_f32)
//   tsum[b,j] = sum_px target[b,j,px]                  (empty-target detection)
//   dots[b,i,j] = sum_px pred[b,i,px]*target[b,j,px]   (V_WMMA_F32_16X16X4_F32)
// Finalize: loss = 5 * mean_b,i min_j [ tsum==0 ? 0 : s/HW - dots/HW ]
// ---------------------------------------------------------------------------

typedef __attribute__((ext_vector_type(2))) float v2f;
typedef __attribute__((ext_vector_type(8))) float v8f;

#define B_      8
#define NP_     8
#define NT_     16
#define HW_     (512 * 512)
#define CHUNK   4096                 // pixels per block
#define NCHUNK  (HW_ / CHUNK)        // 64
#define TILE    512                  // pixels staged in LDS per iteration
#define PITCH   (TILE + 1)           // +1 float pad -> conflict-free strided reads
#define WS_N    (B_*NP_ + B_*NT_ + B_*NP_*NT_)   // 1216 floats of workspace

#define LOG2E_F 1.4426950408889634f
#define LN2_F   0.6931471805599453f

__device__ __forceinline__ float softplus_f(float x) {
    // softplus(x) = max(x,0) + ln2 * log2(1 + 2^(-|x|*log2e))
    // Raw v_exp_f32 / v_log_f32: exp2 arg <= 0 (no overflow), log2 arg in
    // (1,2] (no denorm/inf/zero), so the libm guard sequences (cndmask,
    // ldexp, inf checks) are unnecessary. ~6 VALU ops, 2 of them TRANS32.
    float u = __builtin_amdgcn_exp2f(-LOG2E_F * fabsf(x));   // exp(-|x|)
    float l = __builtin_amdgcn_logf(1.0f + u);               // log2(1+u)
    return fmaf(LN2_F, l, fmaxf(x, 0.0f));
}

__global__ void samloss_zero(float* __restrict__ ws) {
    for (int i = threadIdx.x; i < WS_N; i += blockDim.x) ws[i] = 0.0f;
}

__global__ __launch_bounds__(256) void samloss_partial(
    const float* __restrict__ pred,    // [B, NP, HW]
    const float* __restrict__ tgt,     // [B, NT, HW]
    float* __restrict__ ws)
{
    __shared__ float predT[NP_ * PITCH];   // 16.4 KB
    __shared__ float tgtT [NT_ * PITCH];   // 32.8 KB
    __shared__ float dotsL[NP_ * NT_];     // 0.5 KB

    const int tid   = threadIdx.x;
    const int lane  = tid & 31;
    const int wave  = tid >> 5;            // 8 waves per block (wave32)
    const int chunk = blockIdx.x;          // pixel chunk
    const int b     = blockIdx.y;          // batch

    float* s_g = ws;                             // [B*NP]  softplus sums
    float* t_g = ws + B_*NP_;                    // [B*NT]  target sums
    float* d_g = ws + B_*NP_ + B_*NT_;           // [B*NP*NT] dot sums

    if (tid < NP_ * NT_) dotsL[tid] = 0.0f;

    float sp[NP_];
    float ts[NT_];
#pragma unroll
    for (int r = 0; r < NP_; ++r) sp[r] = 0.0f;
#pragma unroll
    for (int r = 0; r < NT_; ++r) ts[r] = 0.0f;

    v8f acc = {};   // WMMA accumulator: rows 0..7 valid (A rows 8..15 are zero)

    const float* predB = pred + (size_t)b * NP_ * HW_;
    const float* tgtB  = tgt  + (size_t)b * NT_ * HW_;
    const int    px0   = chunk * CHUNK;

    const int half = lane >> 4;        // K half select (A/B operand layout)
    const int m    = lane & 15;        // row (A) / column (B) index

    for (int t0 = 0; t0 < CHUNK; t0 += TILE) {
        __syncthreads();   // tiles safe to overwrite

        // ---- coalesced tile load: pred 8x512, fused softplus accumulate ----
#pragma unroll
        for (int k = 0; k < (NP_ * TILE) / 256; ++k) {       // 16 iters
            int f   = tid + 256 * k;
            int row = f >> 9;                                // == k>>1 (const)
            int col = f & (TILE - 1);
            float v = predB[(size_t)row * HW_ + px0 + t0 + col];
            predT[row * PITCH + col] = v;
            sp[k >> 1] += softplus_f(v);
        }
        // ---- coalesced tile load: target 16x512, fused target-sum ----------
#pragma unroll
        for (int k = 0; k < (NT_ * TILE) / 256; ++k) {       // 32 iters
            int f   = tid + 256 * k;
            int row = f >> 9;
            int col = f & (TILE - 1);
            float v = tgtB[(size_t)row * HW_ + px0 + t0 + col];
            tgtT[row * PITCH + col] = v;
            ts[k >> 1] += v;
        }
        __syncthreads();

        // ---- WMMA phase: wave w reduces K in [w*64, w*64+64) ---------------
        // V_WMMA_F32_16X16X4_F32 operand layout (wave32):
        //   A 16x4 : lanes 0-15 hold M=lane  {K=0,K=1}; lanes 16-31 {K=2,K=3}
        //   B 4x16 : lanes 0-15 hold N=lane  {K=0,K=1}; lanes 16-31 {K=2,K=3}
        const int kbeg = wave * (TILE / 8);
#pragma unroll
        for (int kk = 0; kk < TILE / 8; kk += 4) {
            int kb = kbeg + kk + 2 * half;
            v2f a, bv;
            a.x  = (m < NP_) ? predT[m * PITCH + kb    ] : 0.0f;
            a.y  = (m < NP_) ? predT[m * PITCH + kb + 1] : 0.0f;
            bv.x = tgtT[m * PITCH + kb    ];
            bv.y = tgtT[m * PITCH + kb + 1];
            acc  = __builtin_amdgcn_wmma_f32_16x16x4_f32(
                       false, a, false, bv, (short)0, acc, false, false);
        }
    }

    // ---- fold wave WMMA accumulators into LDS dot matrix -------------------
    // C/D layout: VGPR r, lanes 0-15 -> (M=r, N=lane); lanes 16-31 -> M=r+8
    // (rows 8..15 are identically zero: A rows were zero-padded)
    if (lane < 16) {
#pragma unroll
        for (int r = 0; r < 8; ++r)
            atomicAdd(&dotsL[r * NT_ + lane], acc[r]);
    }

    // ---- wave-reduce softplus / target sums, one atomic per wave per row ---
#pragma unroll
    for (int r = 0; r < NP_; ++r) {
        float v = sp[r];
        for (int off = 16; off > 0; off >>= 1) v += __shfl_down(v, off);
        if (lane == 0) atomicAdd(&s_g[b * NP_ + r], v);
    }
#pragma unroll
    for (int r = 0; r < NT_; ++r) {
        float v = ts[r];
        for (int off = 16; off > 0; off >>= 1) v += __shfl_down(v, off);
        if (lane == 0) atomicAdd(&t_g[b * NT_ + r], v);
    }

    __syncthreads();
    if (tid < NP_ * NT_) atomicAdd(&d_g[b * NP_ * NT_ + tid], dotsL[tid]);
}

__global__ __launch_bounds__(64) void samloss_final(
    const float* __restrict__ ws, float* __restrict__ out)
{
    __shared__ float waveSum[2];
    const int t = threadIdx.x;                 // 64 threads: (b,i) pairs
    const float* s_g = ws;
    const float* t_g = ws + B_*NP_;
    const float* d_g = ws + B_*NP_ + B_*NT_;

    const int   b   = t >> 3, i = t & 7;
    const float inv = 1.0f / (float)HW_;
    const float s   = s_g[b * NP_ + i] * inv;  // mean softplus

    float mn = INFINITY;
#pragma unroll
    for (int j = 0; j < NT_; ++j) {
        // empty targets were skipped in reference -> matrix entry stays 0
        float entry = (t_g[b * NT_ + j] == 0.0f)
                          ? 0.0f
                          : (s - d_g[(b * NP_ + i) * NT_ + j] * inv);
        mn = fminf(mn, entry);
    }

    float v = mn;
    for (int off = 16; off > 0; off >>= 1) v += __shfl_down(v, off);
    if ((t & 31) == 0) waveSum[t >> 5] = v;
    __syncthreads();
    if (t == 0)
        out[0] = 5.0f * (waveSum[0] + waveSum[1]) / (float)(B_ * NP_);
}

extern "C" void kernel_launch(void* const* d_in, const int* in_sizes, int n_in,
                              void* d_out, int out_size, void* d_ws, size_t ws_size,
                              hipStream_t stream) {
    const float* pred = (const float*)d_in[0];   // [8, 8, 512, 512] fp32
    const float* tgt  = (const float*)d_in[1];   // [8, 16, 512, 512] fp32
    float*       ws   = (float*)d_ws;
    float*       out  = (float*)d_out;

    samloss_zero<<<1, 256, 0, stream>>>(ws);
    dim3 grid(NCHUNK, B_);                       // 64 chunks x 8 batches
    samloss_partial<<<grid, 256, 0, stream>>>(pred, tgt, ws);
    samloss_final<<<1, 64, 0, stream>>>(ws, out);
}